// KeplerDiffEq_54391465837051
// MI455X (gfx1250) — compile-verified
//
#include <hip/hip_runtime.h>
#include <hip/hip_bf16.h>
#include <math.h>

typedef float v2f __attribute__((ext_vector_type(2)));
typedef float v8f __attribute__((ext_vector_type(8)));

#define KMU 3.0f
#define KLR 0.01f
#define KTOL 1e-10f
#define KMAX_STEPS 2000

// One wave32 solves the whole problem.
// Lanes replicate the 4 orbits (orbit = lane & 3) through the serial Kepler
// solve; the rotation application (r = C*[x,y], dr = C*[dx,dy] for 4 orbits)
// is done with a single v_wmma_f32_16x16x4_f32.
__global__ __launch_bounds__(32) void KeplerDiffEq_54391465837051_kernel(
    const float* __restrict__ a,
    const float* __restrict__ e,
    const float* __restrict__ inc,
    const float* __restrict__ omg,   // omega (arg of periapsis)
    const float* __restrict__ Omg,   // Omega (RAAN)
    const float* __restrict__ mm,    // mean motion
    const float* __restrict__ Man,   // mean anomaly (scalar)
    const float* __restrict__ x,     // (4,6), we use cols 0,1
    float* __restrict__ out)         // (4,6) = [dr(3), ddr(3)] per orbit
{
    const int lane  = (int)threadIdx.x;      // 0..31
    const int myorb = lane & 3;

    const float M  = Man[0];
    const float el = e[myorb];

    // ---- damped-Newton Kepler solve (matches jax.lax.while_loop semantics) ----
    float E  = M;
    float dE = M;
    int step = 0;
    for (;;) {
        // mean(|dE|) over the 4 orbits; lanes repeat with period 4, so two
        // xor-shuffles give every lane the group-of-4 sum -> uniform branch.
        float ad = fabsf(dE);
        ad += __shfl_xor(ad, 1, 32);
        ad += __shfl_xor(ad, 2, 32);
        const float meanAbs = ad * 0.25f;
        if (!((meanAbs > KTOL) && (step < KMAX_STEPS))) break;
        float s, c;
        sincosf(E, &s, &c);
        dE = (E - el * s - M) / (1.0f - el * c);
        E  = E - KLR * dE;
        ++step;
    }
    // lane q (q = 0..3) now holds E for orbit q.

    // ---- build A fragment (16x4 f32):
    //   rows 0..3 : (x_, y_, 0, 0)      rows 4..7 : (0, 0, dx, dy)
    // layout: lanes 0-15 hold K=0(V0),K=1(V1) of row m=lane;
    //         lanes 16-31 hold K=2(V0),K=3(V1) of row m=lane-16.
    const bool loA  = (lane < 16);
    const bool xrow = loA && (lane < 4);               // rows 0..3, K=0/1
    const bool drow = (!loA) && (lane >= 20) && (lane < 24); // rows 4..7, K=2/3
    const int  qA   = drow ? (lane - 20) : 0;          // orbit for dx/dy rows

    const float Eq  = __shfl(E, qA, 32);
    const float aq  = a[qA], eq = e[qA];
    float sq, cq;
    sincosf(Eq, &sq, &cq);
    const float rcq  = aq * (1.0f - eq * cq);
    const float scal = sqrtf(KMU * aq) / rcq;
    const float dxq  = -sq * scal;
    const float dyq  = sqrtf(1.0f - eq * eq) * cq * scal;

    const int xi = xrow ? lane : 0;                    // clamped x index
    const float xv0 = x[xi * 6 + 0];
    const float xv1 = x[xi * 6 + 1];

    v2f A;
    A[0] = xrow ? xv0 : (drow ? dxq : 0.0f);
    A[1] = xrow ? xv1 : (drow ? dyq : 0.0f);

    // ---- build B fragment (4x16 f32): column n = o*3+comp -> (c1,c2,c1,c2)
    // Both K-halves carry identical values, so the half-wave K split of B
    // cannot change the result.
    const int n      = lane & 15;
    const bool nval  = (n < 12);
    const int  o     = nval ? (n / 3) : 0;
    const int  comp  = n - o * 3;

    float cw, sw, cW, sW, ci, si;
    sincosf(omg[o], &sw, &cw);
    sincosf(Omg[o], &sW, &cW);
    sincosf(inc[o], &si, &ci);
    const float c11 =  cw * cW - sw * sW * ci;
    const float c12 = -sw * cW - cw * sW * ci;
    const float c21 =  cw * sW + sw * cW * ci;
    const float c22 = -sw * sW + cw * cW * ci;
    const float c31 =  sw * si;
    const float c32 =  cw * si;
    const float c1 = (comp == 0) ? c11 : ((comp == 1) ? c21 : c31);
    const float c2 = (comp == 0) ? c12 : ((comp == 1) ? c22 : c32);

    v2f B;
    B[0] = nval ? c1 : 0.0f;
    B[1] = nval ? c2 : 0.0f;

    // ---- one WMMA: D[o][n] = r_comp(o), D[o+4][n] = dr_comp(o) ----
    v8f C = {};
    v8f D = __builtin_amdgcn_wmma_f32_16x16x4_f32(
        /*neg_a=*/false, A, /*neg_b=*/false, B,
        /*c_mod=*/(short)0, C, /*reuse_a=*/false, /*reuse_b=*/false);

    // Readout (lanes 0-15 hold rows 0-7 in VGPRs 0-7, N = lane)
    const float r_val  = (o == 0) ? D[0] : (o == 1) ? D[1] : (o == 2) ? D[2] : D[3];
    const float dr_val = (o == 0) ? D[4] : (o == 1) ? D[5] : (o == 2) ? D[6] : D[7];

    // ---- ddr = acc_scale * r / ||r|| ----
    const float rx = __shfl(r_val, 3 * o + 0, 32);
    const float ry = __shfl(r_val, 3 * o + 1, 32);
    const float rz = __shfl(r_val, 3 * o + 2, 32);
    const float nrm = sqrtf(rx * rx + ry * ry + rz * rz);

    const float Eo  = __shfl(E, o, 32);
    const float ao  = a[o], eo = e[o], mmo = mm[o];
    const float rco = ao * (1.0f - eo * cosf(Eo));
    const float acc = -(mmo * mmo) * ao * ao * ao / (rco * rco);
    const float ddr_val = acc * r_val / nrm;

    if (lane < 12) {
        out[o * 6 + comp]     = dr_val;
        out[o * 6 + 3 + comp] = ddr_val;
    }
}

extern "C" void kernel_launch(void* const* d_in, const int* in_sizes, int n_in,
                              void* d_out, int out_size, void* d_ws, size_t ws_size,
                              hipStream_t stream) {
    (void)in_sizes; (void)n_in; (void)out_size; (void)d_ws; (void)ws_size;
    const float* a   = (const float*)d_in[0];
    const float* e   = (const float*)d_in[1];
    const float* inc = (const float*)d_in[2];
    const float* omg = (const float*)d_in[3];
    const float* Omg = (const float*)d_in[4];
    const float* mm  = (const float*)d_in[5];
    const float* Man = (const float*)d_in[6];
    const float* x   = (const float*)d_in[7];
    float* out = (float*)d_out;

    KeplerDiffEq_54391465837051_kernel<<<dim3(1), dim3(32), 0, stream>>>(
        a, e, inc, omg, Omg, mm, Man, x, out);
}